// ADSCDConv_57166014709877
// MI455X (gfx1250) — compile-verified
//
#include <hip/hip_runtime.h>
#include <math.h>

#define B_   16
#define C_   192
#define R_   48
#define G_   4
#define HIMG 96
#define WIMG 96
#define HW   (HIMG*WIMG)   // 9216
#define W4   (WIMG/4)      // 24

typedef float v2f __attribute__((ext_vector_type(2)));
typedef float v8f __attribute__((ext_vector_type(8)));

__device__ __forceinline__ float wave_reduce_sum(float v) {
#pragma unroll
  for (int off = 16; off > 0; off >>= 1) v += __shfl_xor(v, off, 32);
  return v;
}
__device__ __forceinline__ float wave_reduce_max(float v) {
#pragma unroll
  for (int off = 16; off > 0; off >>= 1) v = fmaxf(v, __shfl_xor(v, off, 32));
  return v;
}

// ---------------------------------------------------------------------------
// Kernel A: per-(b,c) avg, max, and 3x3 average pool (32x32 cells).
// One block per (b,c) image; 256 threads; float4 loads. Each thread owns one
// (row,col4) slot inside every 32x32 cell so the cell index is compile-time.
// ---------------------------------------------------------------------------
__global__ void k_stats(const float* __restrict__ x, float* __restrict__ avg,
                        float* __restrict__ mxv, float* __restrict__ pooled) {
  const int bc  = blockIdx.x;
  const int tid = threadIdx.x;
  const float4* x4 = (const float4*)(x + (size_t)bc * HW);

  const int rloc = tid >> 3;   // 0..31 row inside cell
  const int cloc = tid & 7;    // 0..7 float4-col inside cell

  float cell[9];
#pragma unroll
  for (int i = 0; i < 9; ++i) cell[i] = 0.f;
  float vmax = -INFINITY;

#pragma unroll
  for (int i = 0; i < 9; ++i) {
    const int gr  = (i / 3) * 32 + rloc;
    const int gc4 = (i % 3) * 8 + cloc;
    float4 v = x4[gr * W4 + gc4];
    cell[i] += v.x + v.y + v.z + v.w;
    vmax = fmaxf(vmax, fmaxf(fmaxf(v.x, v.y), fmaxf(v.z, v.w)));
  }

  __shared__ float red[8][10];
  __shared__ float fin[10];
  const int lane = tid & 31, wv = tid >> 5;
#pragma unroll
  for (int i = 0; i < 9; ++i) {
    float s = wave_reduce_sum(cell[i]);
    if (lane == 0) red[wv][i] = s;
  }
  float m = wave_reduce_max(vmax);
  if (lane == 0) red[wv][9] = m;
  __syncthreads();

  if (tid < 10) {
    float a = red[0][tid];
    if (tid == 9) { for (int w = 1; w < 8; ++w) a = fmaxf(a, red[w][9]); }
    else          { for (int w = 1; w < 8; ++w) a += red[w][tid]; }
    fin[tid] = a;
  }
  __syncthreads();

  if (tid < 9) pooled[(size_t)bc * 9 + tid] = fin[tid] * (1.f / 1024.f);
  if (tid == 0) {
    float t = 0.f;
    for (int i = 0; i < 9; ++i) t += fin[i];
    avg[bc] = t * (1.f / 9216.f);
  }
  if (tid == 1) mxv[bc] = fin[9];
}

// ---------------------------------------------------------------------------
// Kernel B: theta = sigmoid( relu(avg@W1^T)@W2^T + relu(mx@W1^T)@W2^T )
// Exact f32 math via V_WMMA_F32_16X16X4_F32 (M=16=B).
// Phase 1: H = relu([avg|mx] @ W1^T) summed into LDS (16x48).
// Phase 2: T = H @ W2^T (16x192), sigmoid, store.
// 1 block, 128 threads = 4 waves; all wave branches are wave-uniform so
// EXEC is all-ones at every WMMA (ISA requirement).
// Fragment layouts per ISA 7.12.2:
//   A (16x4 f32):  lane<16 holds A[l][k0..k0+1], lane>=16 holds A[l-16][k0+2..3]
//   B (4x16 f32):  mirror: lane holds B[k0+2*hi + {0,1}][lane%16]
//   C/D (16x16):   element (m,n) in vgpr m%8, lane n + 16*(m/8)
// ---------------------------------------------------------------------------
__global__ void k_theta(const float* __restrict__ avg, const float* __restrict__ mxv,
                        const float* __restrict__ cam_w1, const float* __restrict__ cam_w2,
                        float* __restrict__ theta) {
  __shared__ float Hs[16 * 48];
  const int tid  = threadIdx.x;
  const int lane = tid & 31;
  const int wv   = tid >> 5;   // 0..3
  const int hi   = lane >> 4;  // 0/1
  const int lm   = lane & 15;

  for (int j = tid; j < 16 * 48; j += blockDim.x) Hs[j] = 0.f;
  __syncthreads();

  // Phase 1: six 16x16 output tiles: (src in {avg,mx}) x (ntile 0..2 over R=48)
  for (int t = wv; t < 6; t += 4) {
    const float* A = (t < 3) ? avg : mxv;     // [16 x 192] row-major
    const int nt = t % 3;
    v8f acc = {};
    for (int kt = 0; kt < 48; ++kt) {         // K = 192 in steps of 4
      const int k0 = kt * 4 + 2 * hi;
      v2f a, b;
      a.x = A[lm * C_ + k0];
      a.y = A[lm * C_ + k0 + 1];
      // B1[k][n] = cam_w1[nt*16+n][k]  (cam_w1 is [R=48 x C=192] row-major)
      b.x = cam_w1[(nt * 16 + lm) * C_ + k0];
      b.y = cam_w1[(nt * 16 + lm) * C_ + k0 + 1];
      acc = __builtin_amdgcn_wmma_f32_16x16x4_f32(false, a, false, b,
                                                  (short)0, acc, false, false);
    }
#pragma unroll
    for (int v = 0; v < 8; ++v) {
      const int mrow = v + 8 * hi;
      atomicAdd(&Hs[mrow * 48 + nt * 16 + lm], fmaxf(acc[v], 0.f));
    }
  }
  __syncthreads();

  // Phase 2: twelve 16x16 output tiles over C=192; K = 48 in steps of 4
  for (int t = wv * 3; t < wv * 3 + 3; ++t) {
    v8f acc = {};
    for (int kt = 0; kt < 12; ++kt) {
      const int k0 = kt * 4 + 2 * hi;
      v2f a, b;
      a.x = Hs[lm * 48 + k0];
      a.y = Hs[lm * 48 + k0 + 1];
      // B2[k][n] = cam_w2[t*16+n][k]  (cam_w2 is [C=192 x R=48] row-major)
      b.x = cam_w2[(t * 16 + lm) * R_ + k0];
      b.y = cam_w2[(t * 16 + lm) * R_ + k0 + 1];
      acc = __builtin_amdgcn_wmma_f32_16x16x4_f32(false, a, false, b,
                                                  (short)0, acc, false, false);
    }
#pragma unroll
    for (int v = 0; v < 8; ++v) {
      const int mrow = v + 8 * hi;
      const float z = acc[v];
      theta[mrow * C_ + t * 16 + lm] = 1.f / (1.f + __expf(-z));
    }
  }
}

// ---------------------------------------------------------------------------
// Kernel C: per-(b,c) adaptive 3x3 kernel.
// One block per batch b; 192 threads (one per channel in phase 2).
// ---------------------------------------------------------------------------
__global__ void k_weight(const float* __restrict__ pooled, const float* __restrict__ theta,
                         const float* __restrict__ proj_w1, const float* __restrict__ bn_gamma,
                         const float* __restrict__ bn_beta, const float* __restrict__ proj_w2,
                         const float* __restrict__ adk, float* __restrict__ wker) {
  const int b   = blockIdx.x;
  const int tid = threadIdx.x;   // 192
  __shared__ float pl[C_ * 9];
  __shared__ float hs[R_ * 9];

  for (int j = tid; j < C_ * 9; j += blockDim.x)
    pl[j] = pooled[(size_t)b * C_ * 9 + j];
  __syncthreads();

  const float invs = rsqrtf(1.f + 1e-5f);
  for (int j = tid; j < R_ * 9; j += blockDim.x) {
    const int r = j / 9, cell = j % 9;
    float acc = 0.f;
    for (int c = 0; c < C_; ++c) acc += pl[c * 9 + cell] * proj_w1[r * C_ + c];
    hs[j] = fmaxf(acc * (bn_gamma[r] * invs) + bn_beta[r], 0.f);
  }
  __syncthreads();

  const int c = tid;
  float sacc[G_][9];
#pragma unroll
  for (int g = 0; g < G_; ++g)
#pragma unroll
    for (int cl = 0; cl < 9; ++cl) sacc[g][cl] = 0.f;

  for (int r = 0; r < R_; ++r) {
    float w2v[G_];
#pragma unroll
    for (int g = 0; g < G_; ++g) w2v[g] = proj_w2[((size_t)(g * C_ + c)) * R_ + r];
#pragma unroll
    for (int cl = 0; cl < 9; ++cl) {
      const float hv = hs[r * 9 + cl];
#pragma unroll
      for (int g = 0; g < G_; ++g) sacc[g][cl] += w2v[g] * hv;
    }
  }

  float wloc[9];
  float wsum = 0.f;
#pragma unroll
  for (int cl = 0; cl < 9; ++cl) {
    float mg = sacc[0][cl];
#pragma unroll
    for (int g = 1; g < G_; ++g) mg = fmaxf(mg, sacc[g][cl]);
    float e[G_], tot = 0.f;
#pragma unroll
    for (int g = 0; g < G_; ++g) { e[g] = __expf(sacc[g][cl] - mg); tot += e[g]; }
    const float rinv = 1.f / tot;
    float wvv = 0.f;
#pragma unroll
    for (int g = 0; g < G_; ++g)
      wvv += (e[g] * rinv) * adk[((size_t)(g * C_ + c)) * 9 + cl];
    wloc[cl] = wvv;
    wsum += wvv;
  }

  const float th = theta[b * C_ + c];
  const float delta = wsum - th * wloc[4];
  wloc[4] -= delta;
#pragma unroll
  for (int cl = 0; cl < 9; ++cl)
    wker[((size_t)(b * C_ + c)) * 9 + cl] = wloc[cl];
}

// ---------------------------------------------------------------------------
// Kernel D: depthwise 3x3 conv, same padding. Bandwidth-critical kernel.
// Block = (bc, strip of 32 rows); LDS holds 34 rows x 96 floats (halo, zero
// padded). float4 global loads and stores, fully coalesced.
// ---------------------------------------------------------------------------
__global__ void k_conv(const float* __restrict__ x, const float* __restrict__ wker,
                       float* __restrict__ out) {
  const int bc    = blockIdx.x;
  const int strip = blockIdx.y;   // 0..2
  const int tid   = threadIdx.x;  // 256
  __shared__ float s[34 * 96];

  const float4* x4 = (const float4*)(x + (size_t)bc * HW);
  for (int j = tid; j < 34 * W4; j += 256) {
    const int r = j / W4, c4 = j % W4;
    const int gr = strip * 32 + r - 1;
    float4 v = {0.f, 0.f, 0.f, 0.f};
    if (gr >= 0 && gr < HIMG) v = x4[gr * W4 + c4];
    ((float4*)(s + r * 96))[c4] = v;
  }

  float w[9];
#pragma unroll
  for (int i = 0; i < 9; ++i) w[i] = wker[(size_t)bc * 9 + i];
  __syncthreads();

  float4* o4 = (float4*)(out + (size_t)bc * HW);
#pragma unroll
  for (int i = 0; i < 3; ++i) {
    const int j  = tid + i * 256;  // 0..767
    const int r  = j / W4;         // 0..31 output row in strip
    const int c4 = j % W4;
    const float* r0 = s + r * 96;        // y-1
    const float* r1 = r0 + 96;           // y
    const float* r2 = r1 + 96;           // y+1
    const int xb = c4 * 4;
    const bool lft = (c4 == 0), rgt = (c4 == W4 - 1);

    const float tm = lft ? 0.f : r0[xb - 1];
    const float t0 = r0[xb], t1 = r0[xb + 1], t2 = r0[xb + 2], t3 = r0[xb + 3];
    const float tp = rgt ? 0.f : r0[xb + 4];
    const float mm = lft ? 0.f : r1[xb - 1];
    const float m0 = r1[xb], m1 = r1[xb + 1], m2 = r1[xb + 2], m3 = r1[xb + 3];
    const float mp = rgt ? 0.f : r1[xb + 4];
    const float bm = lft ? 0.f : r2[xb - 1];
    const float b0 = r2[xb], b1 = r2[xb + 1], b2 = r2[xb + 2], b3 = r2[xb + 3];
    const float bp = rgt ? 0.f : r2[xb + 4];

    float4 o;
    o.x = w[0]*tm + w[1]*t0 + w[2]*t1 + w[3]*mm + w[4]*m0 + w[5]*m1 + w[6]*bm + w[7]*b0 + w[8]*b1;
    o.y = w[0]*t0 + w[1]*t1 + w[2]*t2 + w[3]*m0 + w[4]*m1 + w[5]*m2 + w[6]*b0 + w[7]*b1 + w[8]*b2;
    o.z = w[0]*t1 + w[1]*t2 + w[2]*t3 + w[3]*m1 + w[4]*m2 + w[5]*m3 + w[6]*b1 + w[7]*b2 + w[8]*b3;
    o.w = w[0]*t2 + w[1]*t3 + w[2]*tp + w[3]*m2 + w[4]*m3 + w[5]*mp + w[6]*b2 + w[7]*b3 + w[8]*bp;

    o4[(strip * 32 + r) * W4 + c4] = o;
  }
}

// ---------------------------------------------------------------------------
extern "C" void kernel_launch(void* const* d_in, const int* in_sizes, int n_in,
                              void* d_out, int out_size, void* d_ws, size_t ws_size,
                              hipStream_t stream) {
  const float* x        = (const float*)d_in[0];
  const float* cam_w1   = (const float*)d_in[1];
  const float* cam_w2   = (const float*)d_in[2];
  const float* proj_w1  = (const float*)d_in[3];
  const float* bn_gamma = (const float*)d_in[4];
  const float* bn_beta  = (const float*)d_in[5];
  const float* proj_w2  = (const float*)d_in[6];
  const float* adk      = (const float*)d_in[7];
  float* out = (float*)d_out;

  float* ws     = (float*)d_ws;
  float* avg    = ws;                    // B*C
  float* mxv    = avg + B_ * C_;         // B*C
  float* pooled = mxv + B_ * C_;         // B*C*9
  float* theta  = pooled + B_ * C_ * 9;  // B*C
  float* wker   = theta + B_ * C_;       // B*C*9

  k_stats <<<dim3(B_ * C_),    256, 0, stream>>>(x, avg, mxv, pooled);
  k_theta <<<1,                128, 0, stream>>>(avg, mxv, cam_w1, cam_w2, theta);
  k_weight<<<dim3(B_),         192, 0, stream>>>(pooled, theta, proj_w1, bn_gamma,
                                                 bn_beta, proj_w2, adk, wker);
  k_conv  <<<dim3(B_ * C_, 3), 256, 0, stream>>>(x, wker, out);
}